// ProteinGCN_32014686224487
// MI455X (gfx1250) — compile-verified
//
#include <hip/hip_runtime.h>
#include <hip/hip_bf16.h>

typedef float v2f __attribute__((ext_vector_type(2)));
typedef float v8f __attribute__((ext_vector_type(8)));

// ---------------------------------------------------------------------------
// Utility: zero a float region (graph-capture-safe, no memset dependency)
// ---------------------------------------------------------------------------
__global__ void zero_f32(float* __restrict__ p, long long n) {
    long long i = (long long)blockIdx.x * blockDim.x + threadIdx.x;
    if (i < n) p[i] = 0.0f;
}

// ---------------------------------------------------------------------------
// Degree counting: deg_out over src, deg_in over dst (float atomics)
// ---------------------------------------------------------------------------
__global__ void deg_count(const int* __restrict__ src, const int* __restrict__ dst,
                          float* __restrict__ dsrc, float* __restrict__ ddst, int E) {
    int i = blockIdx.x * blockDim.x + threadIdx.x;
    if (i < E) {
        atomicAdd(&dsrc[src[i]], 1.0f);
        atomicAdd(&ddst[dst[i]], 1.0f);
    }
}

// deg -> rsqrt(max(deg, 1))   (matches clip(deg,1)^-0.5)
__global__ void deg_finalize(float* __restrict__ deg, int n) {
    int i = blockIdx.x * blockDim.x + threadIdx.x;
    if (i < n) deg[i] = rsqrtf(fmaxf(deg[i], 1.0f));
}

// ---------------------------------------------------------------------------
// Edge scatter: agg[dst] += h[src] * deg_out^-1/2[src]   (D = 4*D4 channels)
// One thread handles 4 channels of one edge (float4 gather, 4 float atomics).
// All buffers are L2-resident on MI455X (192MB L2), so these atomics stay on-chip.
// ---------------------------------------------------------------------------
__global__ void scatter_agg(const float* __restrict__ h, const int* __restrict__ src,
                            const int* __restrict__ dst, const float* __restrict__ dsrc,
                            float* __restrict__ agg, int E, int D4) {
    long long gid = (long long)blockIdx.x * blockDim.x + threadIdx.x;
    long long tot = (long long)E * D4;
    if (gid >= tot) return;
    int e  = (int)(gid / D4);
    int c4 = (int)(gid % D4) * 4;
    int s = src[e], d = dst[e];
    float sc = dsrc[s];
    const float4 v = *(const float4*)(h + (size_t)s * (D4 * 4) + c4);
    float* o = agg + (size_t)d * (D4 * 4) + c4;
    atomicAdd(o + 0, v.x * sc);
    atomicAdd(o + 1, v.y * sc);
    atomicAdd(o + 2, v.z * sc);
    atomicAdd(o + 3, v.w * sc);
}

// ---------------------------------------------------------------------------
// WMMA f32 16x16x4 fragments (wave32, one wave per 16x16 tile).
// A 16x4 layout: lanes 0-15 -> M=0..15 ; VGPR0 holds K=0 (lo half) / K=2 (hi),
//                VGPR1 holds K=1 / K=3  => lane reads (m, kbase + 2*half + {0,1}).
// B 4x16 mirrored: lane reads (kbase + 2*half + {0,1}, n).
// C/D: VGPR r -> row = 8*half + r, col = lane&15.
// K and Nout are compile-time so A/B loads use immediate offsets from one base.
// ---------------------------------------------------------------------------

// out[n, :] = sum_t (agg_t[n,:] * ddst_t[n]) @ W_t  + (b0+b1+b2)
template <int K, int Nout>
__global__ void combine_wmma(const float* __restrict__ agg0, const float* __restrict__ agg1,
                             const float* __restrict__ agg2,
                             const float* __restrict__ dd0, const float* __restrict__ dd1,
                             const float* __restrict__ dd2,
                             const float* __restrict__ W0, const float* __restrict__ W1,
                             const float* __restrict__ W2,
                             const float* __restrict__ b0, const float* __restrict__ b1,
                             const float* __restrict__ b2,
                             float* __restrict__ out) {
    const int lane  = threadIdx.x;          // 0..31, full wave
    const int half  = lane >> 4;            // 0 | 1
    const int l16   = lane & 15;
    const int tileN = blockIdx.x;
    const int tileM = blockIdx.y;
    const int m  = tileM * 16 + l16;        // A row for this lane
    const int nn = tileN * 16 + l16;        // B/D column for this lane

    // bias depends only on this lane's column -> fold into accumulator init
    const float bias = b0[nn] + b1[nn] + b2[nn];
    v8f acc = {bias, bias, bias, bias, bias, bias, bias, bias};

    const float* aggs[3] = {agg0, agg1, agg2};
    const float* dds[3]  = {dd0, dd1, dd2};
    const float* Ws[3]   = {W0, W1, W2};

#pragma unroll
    for (int t = 0; t < 3; ++t) {
        // per-lane base pointers; all loop offsets below are compile-time imms
        const float* __restrict__ Arow = aggs[t] + (size_t)m * K + half * 2;
        const float* __restrict__ Wcol = Ws[t] + (size_t)(half * 2) * Nout + nn;
        const float scale = dds[t][m];
#pragma unroll
        for (int k = 0; k < K; k += 4) {
            v2f a = *(const v2f*)(Arow + k);      // 8B-aligned contiguous pair
            a.x *= scale;
            a.y *= scale;
            v2f b;
            b.x = Wcol[(size_t)k * Nout];
            b.y = Wcol[(size_t)(k + 1) * Nout];
            acc = __builtin_amdgcn_wmma_f32_16x16x4_f32(
                false, a, false, b, (short)0, acc, false, false);
        }
    }

#pragma unroll
    for (int r = 0; r < 8; ++r) {
        int row = tileM * 16 + half * 8 + r;
        out[(size_t)row * Nout + nn] = acc[r];
    }
}

// out = relu(h * scale + shift) @ W + bias   (BN affine + ReLU fused into A load)
template <int K, int Nout>
__global__ void fc_wmma(const float* __restrict__ h, const float* __restrict__ scale,
                        const float* __restrict__ shift, const float* __restrict__ W,
                        const float* __restrict__ bias, float* __restrict__ out) {
    const int lane  = threadIdx.x;
    const int half  = lane >> 4;
    const int l16   = lane & 15;
    const int tileN = blockIdx.x;
    const int tileM = blockIdx.y;
    const int m  = tileM * 16 + l16;
    const int nn = tileN * 16 + l16;

    const float bv = bias[nn];
    v8f acc = {bv, bv, bv, bv, bv, bv, bv, bv};

    const float* __restrict__ Arow = h + (size_t)m * K + half * 2;
    const float* __restrict__ Wcol = W + (size_t)(half * 2) * Nout + nn;
    const float* __restrict__ sc2  = scale + half * 2;
    const float* __restrict__ sh2  = shift + half * 2;

#pragma unroll
    for (int k = 0; k < K; k += 4) {
        v2f a = *(const v2f*)(Arow + k);
        a.x = fmaxf(a.x * sc2[k + 0] + sh2[k + 0], 0.0f);
        a.y = fmaxf(a.y * sc2[k + 1] + sh2[k + 1], 0.0f);
        v2f b;
        b.x = Wcol[(size_t)k * Nout];
        b.y = Wcol[(size_t)(k + 1) * Nout];
        acc = __builtin_amdgcn_wmma_f32_16x16x4_f32(
            false, a, false, b, (short)0, acc, false, false);
    }

#pragma unroll
    for (int r = 0; r < 8; ++r) {
        int row = tileM * 16 + half * 8 + r;
        out[(size_t)row * Nout + nn] = acc[r];
    }
}

// ---------------------------------------------------------------------------
// BatchNorm statistics: one block per channel (no atomics needed)
// ---------------------------------------------------------------------------
__global__ void bn_stats(const float* __restrict__ h, float* __restrict__ sum,
                         float* __restrict__ sumsq, int M, int C) {
    const int c = blockIdx.x;
    float s = 0.f, q = 0.f;
    for (int r = threadIdx.x; r < M; r += blockDim.x) {
        float v = h[(size_t)r * C + c];
        s += v;
        q += v * v;
    }
    __shared__ float ss[256], sq[256];
    ss[threadIdx.x] = s;
    sq[threadIdx.x] = q;
    __syncthreads();
    for (int o = blockDim.x / 2; o > 0; o >>= 1) {
        if (threadIdx.x < o) {
            ss[threadIdx.x] += ss[threadIdx.x + o];
            sq[threadIdx.x] += sq[threadIdx.x + o];
        }
        __syncthreads();
    }
    if (threadIdx.x == 0) { sum[c] = ss[0]; sumsq[c] = sq[0]; }
}

__global__ void bn_finalize(const float* __restrict__ sum, const float* __restrict__ sumsq,
                            const float* __restrict__ g, const float* __restrict__ be,
                            float* __restrict__ scale, float* __restrict__ shift,
                            int M, int C) {
    int c = blockIdx.x * blockDim.x + threadIdx.x;
    if (c < C) {
        float mean = sum[c] / (float)M;
        float var  = sumsq[c] / (float)M - mean * mean;
        float sc   = g[c] * rsqrtf(var + 1e-5f);
        scale[c] = sc;
        shift[c] = be[c] - mean * sc;
    }
}

// ---------------------------------------------------------------------------
// Launch
// ---------------------------------------------------------------------------
extern "C" void kernel_launch(void* const* d_in, const int* in_sizes, int n_in,
                              void* d_out, int out_size, void* d_ws, size_t ws_size,
                              hipStream_t stream) {
    const float* x       = (const float*)d_in[0];
    const int*   srcK    = (const int*)d_in[1];
    const int*   dstK    = (const int*)d_in[2];
    const int*   srcS    = (const int*)d_in[3];
    const int*   dstS    = (const int*)d_in[4];
    const int*   srcQ    = (const int*)d_in[5];
    const int*   dstQ    = (const int*)d_in[6];
    const float* w0_knn  = (const float*)d_in[7];
    const float* b0_knn  = (const float*)d_in[8];
    const float* w0_sph  = (const float*)d_in[9];
    const float* b0_sph  = (const float*)d_in[10];
    const float* w0_seq  = (const float*)d_in[11];
    const float* b0_seq  = (const float*)d_in[12];
    const float* bn0_g   = (const float*)d_in[13];
    const float* bn0_b   = (const float*)d_in[14];
    const float* fc0_w   = (const float*)d_in[15];
    const float* fc0_b   = (const float*)d_in[16];
    const float* w1_knn  = (const float*)d_in[17];
    const float* b1_knn  = (const float*)d_in[18];
    const float* w1_sph  = (const float*)d_in[19];
    const float* b1_sph  = (const float*)d_in[20];
    const float* w1_seq  = (const float*)d_in[21];
    const float* b1_seq  = (const float*)d_in[22];
    const float* bn1_g   = (const float*)d_in[23];
    const float* bn1_b   = (const float*)d_in[24];
    const float* fc1_w   = (const float*)d_in[25];
    const float* fc1_b   = (const float*)d_in[26];

    const int N  = in_sizes[0] / 64;   // 64000 nodes
    const int EK = in_sizes[1];
    const int ES = in_sizes[3];
    const int EQ = in_sizes[5];
    const size_t n = (size_t)N;

    // Workspace layout (floats): all L2-resident on MI455X (192MB global L2)
    float* ws    = (float*)d_ws;
    float* deg   = ws;                  // 6n: dsrcK,ddstK,dsrcS,ddstS,dsrcQ,ddstQ
    float* dsrcK = deg + 0 * n;
    float* ddstK = deg + 1 * n;
    float* dsrcS = deg + 2 * n;
    float* ddstS = deg + 3 * n;
    float* dsrcQ = deg + 4 * n;
    float* ddstQ = deg + 5 * n;
    float* agg0  = deg  + 6 * n;        // 96n each (max D across layers)
    float* agg1  = agg0 + 96 * n;
    float* agg2  = agg1 + 96 * n;
    float* hsum  = agg2 + 96 * n;       // 128n (max)
    float* hmid  = hsum + 128 * n;      // 96n
    float* stats = hmid + 96 * n;       // 512: sum | sumsq | scale | shift
    float* bnsum = stats;
    float* bnsq  = stats + 128;
    float* scale = stats + 256;
    float* shift = stats + 384;

    const int TB = 256;
    auto blk = [&](long long tot) { return (unsigned)((tot + TB - 1) / TB); };

    // ---- degrees (recomputed each call; deterministic) ----
    zero_f32<<<blk(6 * (long long)N), TB, 0, stream>>>(deg, 6 * (long long)N);
    deg_count<<<blk(EK), TB, 0, stream>>>(srcK, dstK, dsrcK, ddstK, EK);
    deg_count<<<blk(ES), TB, 0, stream>>>(srcS, dstS, dsrcS, ddstS, ES);
    deg_count<<<blk(EQ), TB, 0, stream>>>(srcQ, dstQ, dsrcQ, ddstQ, EQ);
    deg_finalize<<<blk(6 * (long long)N), TB, 0, stream>>>(deg, 6 * N);

    // ================= layer 0 (64 -> 96) =================
    zero_f32<<<blk(3 * 96 * (long long)N), TB, 0, stream>>>(agg0, 3 * 96 * (long long)N);
    scatter_agg<<<blk((long long)EK * 16), TB, 0, stream>>>(x, srcK, dstK, dsrcK, agg0, EK, 16);
    scatter_agg<<<blk((long long)ES * 16), TB, 0, stream>>>(x, srcS, dstS, dsrcS, agg1, ES, 16);
    scatter_agg<<<blk((long long)EQ * 16), TB, 0, stream>>>(x, srcQ, dstQ, dsrcQ, agg2, EQ, 16);

    combine_wmma<64, 96><<<dim3(96 / 16, N / 16), 32, 0, stream>>>(
        agg0, agg1, agg2, ddstK, ddstS, ddstQ,
        w0_knn, w0_sph, w0_seq, b0_knn, b0_sph, b0_seq, hsum);

    bn_stats<<<96, 256, 0, stream>>>(hsum, bnsum, bnsq, N, 96);
    bn_finalize<<<1, 128, 0, stream>>>(bnsum, bnsq, bn0_g, bn0_b, scale, shift, N, 96);

    fc_wmma<96, 96><<<dim3(96 / 16, N / 16), 32, 0, stream>>>(
        hsum, scale, shift, fc0_w, fc0_b, hmid);

    // ================= layer 1 (96 -> 128) =================
    zero_f32<<<blk(3 * 96 * (long long)N), TB, 0, stream>>>(agg0, 3 * 96 * (long long)N);
    scatter_agg<<<blk((long long)EK * 24), TB, 0, stream>>>(hmid, srcK, dstK, dsrcK, agg0, EK, 24);
    scatter_agg<<<blk((long long)ES * 24), TB, 0, stream>>>(hmid, srcS, dstS, dsrcS, agg1, ES, 24);
    scatter_agg<<<blk((long long)EQ * 24), TB, 0, stream>>>(hmid, srcQ, dstQ, dsrcQ, agg2, EQ, 24);

    combine_wmma<96, 128><<<dim3(128 / 16, N / 16), 32, 0, stream>>>(
        agg0, agg1, agg2, ddstK, ddstS, ddstQ,
        w1_knn, w1_sph, w1_seq, b1_knn, b1_sph, b1_seq, hsum);

    bn_stats<<<128, 256, 0, stream>>>(hsum, bnsum, bnsq, N, 128);
    bn_finalize<<<1, 128, 0, stream>>>(bnsum, bnsq, bn1_g, bn1_b, scale, shift, N, 128);

    fc_wmma<128, 128><<<dim3(128 / 16, N / 16), 32, 0, stream>>>(
        hsum, scale, shift, fc1_w, fc1_b, (float*)d_out);
}